// DECOMP_13511967113408
// MI455X (gfx1250) — compile-verified
//
#include <hip/hip_runtime.h>
#include <cstdint>

#define DEMA_ALPHA 0.3f
#define DEMA_BETA  0.3f

namespace {
constexpr int kB = 32;
constexpr int kT = 2048;
constexpr int kF = 512;
constexpr int kChunk = 256;   // timesteps written per block
constexpr int kWarm  = 128;   // warm-up halo; state error decays ~0.8367^128 ~ 1e-10
constexpr int kFT    = 256;   // features per block (== blockDim.x)
constexpr int kSlab  = 16;    // timesteps staged per LDS slab
constexpr int kRing  = 4;     // LDS ring depth: 3 slabs (48 KB) in flight
constexpr int kSlabFloats = kSlab * kFT;  // 4096 floats = 16 KB per buffer
}

// Pointer/vector types matching the async builtin's signature:
//   param0: int __vector(4) __device__ *   (AS1, global)
//   param1: int __vector(4) __shared__ *   (AS3, LDS)
typedef int v4i_vec __attribute__((vector_size(16)));
typedef __attribute__((address_space(1))) v4i_vec* gbl_v4i_p;
typedef __attribute__((address_space(3))) v4i_vec* lds_v4i_p;

// ---- CDNA5 async global->LDS copy (ASYNCcnt-tracked), 16 bytes per lane ----
__device__ __forceinline__ void async_ld_b128(const float* g, unsigned lds_off) {
#if __has_builtin(__builtin_amdgcn_global_load_async_to_lds_b128)
  __builtin_amdgcn_global_load_async_to_lds_b128(
      (gbl_v4i_p)(uintptr_t)g, (lds_v4i_p)lds_off, /*offset=*/0, /*cpol=*/0);
#else
  asm volatile("global_load_async_to_lds_b128 %0, %1, off"
               :: "v"(lds_off), "v"(g) : "memory");
#endif
}

template <int N>
__device__ __forceinline__ void wait_asynccnt() {
#if __has_builtin(__builtin_amdgcn_s_wait_asynccnt)
  __builtin_amdgcn_s_wait_asynccnt(N);
#else
  asm volatile("s_wait_asynccnt %0" :: "i"(N) : "memory");
#endif
}

__device__ __forceinline__ void dema_step(float v, float& s, float& bb) {
  const float sp = s;
  s  = fmaf(DEMA_ALPHA, v, (1.0f - DEMA_ALPHA) * (sp + bb));
  bb = fmaf(DEMA_BETA, s - sp, (1.0f - DEMA_BETA) * bb);
}

// Consume one 16-step slab from LDS. FIRST: peel the s0/b0 init steps.
// WRITE: emit ma/res (warm-up slabs skip this). maP/resP advance by kF/step.
template <bool WRITE, bool FIRST>
__device__ __forceinline__ void do_slab(const float* __restrict__ buf, int tid,
                                        float& s, float& bb,
                                        float*& maP, float*& resP) {
  if (FIRST) {
    const float v0 = buf[tid];            // s0 = x[t0], ma[t0] = s0, res = 0
    s = v0;
    if (WRITE) {
      __builtin_nontemporal_store(s, maP);
      __builtin_nontemporal_store(0.0f, resP);
      maP += kF; resP += kF;
    }
    const float v1 = buf[kFT + tid];      // b0 = x[t0+1] - s0, then step
    bb = v1 - s;
    dema_step(v1, s, bb);
    if (WRITE) {
      __builtin_nontemporal_store(s, maP);
      __builtin_nontemporal_store(v1 - s, resP);
      maP += kF; resP += kF;
    }
  }
#pragma unroll
  for (int k = (FIRST ? 2 : 0); k < kSlab; ++k) {
    const float v = buf[k * kFT + tid];   // 32 consecutive dwords per wave
    dema_step(v, s, bb);
    if (WRITE) {
      __builtin_nontemporal_store(s, maP);
      __builtin_nontemporal_store(v - s, resP);
      maP += kF; resP += kF;
    }
  }
}

__global__ __launch_bounds__(kFT) void dema_kernel(const float* __restrict__ x,
                                                   float* __restrict__ res,
                                                   float* __restrict__ ma) {
  __shared__ float lds[kRing][kSlabFloats];   // 64 KB ring

  const int tid   = threadIdx.x;
  const int fbase = blockIdx.x * kFT;   // 0 or 256
  const int chunk = blockIdx.y;         // 0..7
  const int batch = blockIdx.z;         // 0..31

  const int wstart = chunk * kChunk;                      // first written t
  const int t0     = (chunk == 0) ? 0 : (wstart - kWarm); // warm-up start
  const int nslab  = (wstart + kChunk - t0) / kSlab;      // 16 or 24
  const int warmSl = (chunk == 0) ? 0 : (kWarm / kSlab);  // 0 or 8

  const float* xb = x + (size_t)batch * kT * kF + fbase;
  float* maP  = ma  + (size_t)batch * kT * kF + fbase + (size_t)wstart * kF + tid;
  float* resP = res + (size_t)batch * kT * kF + fbase + (size_t)wstart * kF + tid;

  unsigned ldsb[kRing];
#pragma unroll
  for (int r = 0; r < kRing; ++r) ldsb[r] = (unsigned)(uintptr_t)&lds[r][0];

  // Cooperative async load of one 16-step x-slab: 1024 x 16B granules,
  // 4 per thread. Lanes cover contiguous 512B runs in both global and LDS.
  auto issue_slab = [&](int si) {
    const int trow0 = t0 + si * kSlab;
    const unsigned base = ldsb[si & (kRing - 1)];
#pragma unroll
    for (int j = 0; j < 4; ++j) {
      const int slot = tid + j * kFT;     // 0..1023
      const int row  = slot >> 6;         // 64 granules per 1 KB row
      const int col  = (slot & 63) << 2;  // float index within row
      async_ld_b128(xb + (size_t)(trow0 + row) * kF + col,
                    base + (unsigned)slot * 16u);
    }
  };

  // Prologue: prime 3 slabs of the ring (12 async ops, 48 KB in flight).
  issue_slab(0); issue_slab(1); issue_slab(2);

  float s = 0.0f, bb = 0.0f;
  for (int si = 0; si < nslab; ++si) {
    // In-order completion: slab si has landed for THIS wave when at most the
    // newer slabs' loads (4 each) remain outstanding.
    const int rem = nslab - 1 - si;
    if (rem >= 2)      wait_asynccnt<8>();
    else if (rem == 1) wait_asynccnt<4>();
    else               wait_asynccnt<0>();
    __syncthreads();   // slab si visible everywhere; compute si-1 done everywhere

    // Refill the buffer freed at si-1 (safe: barrier above covers its readers).
    if (si + 3 < nslab) issue_slab(si + 3);

    const float* buf = &lds[si & (kRing - 1)][0];
    if (si == 0) {
      if (warmSl == 0) do_slab<true,  true >(buf, tid, s, bb, maP, resP);
      else             do_slab<false, true >(buf, tid, s, bb, maP, resP);
    } else if (si < warmSl) {
      do_slab<false, false>(buf, tid, s, bb, maP, resP);
    } else {
      do_slab<true,  false>(buf, tid, s, bb, maP, resP);
    }
  }
}

extern "C" void kernel_launch(void* const* d_in, const int* in_sizes, int n_in,
                              void* d_out, int out_size, void* d_ws, size_t ws_size,
                              hipStream_t stream) {
  (void)in_sizes; (void)n_in; (void)out_size; (void)d_ws; (void)ws_size;
  const float* x = (const float*)d_in[0];
  float* res = (float*)d_out;                       // output 0: x - ma
  float* ma  = res + (size_t)kB * kT * kF;          // output 1: ma
  dim3 grid(kF / kFT, kT / kChunk, kB);             // (2, 8, 32) = 512 blocks
  dema_kernel<<<grid, dim3(kFT), 0, stream>>>(x, res, ma);
}